// _HydrophobicPairs_58256936403302
// MI455X (gfx1250) — compile-verified
//
#include <hip/hip_runtime.h>

#define THREADS 256
#define LTILE   512
#define LROW    8192            // L (compile-time; matches reference)
#define KNB     64              // K (compile-time; matches reference)
#define TILES_PER_B (LROW / LTILE)   // 16

typedef float v4f __attribute__((ext_vector_type(4)));
typedef int   v4i __attribute__((ext_vector_type(4)));

__device__ __forceinline__ float hp_elem(float rv, int jv, const float* __restrict__ hj,
                                         float md, float r_peak, float inv2s2) {
    int js = jv < 0 ? 0 : (jv > (LROW - 1) ? (LROW - 1) : jv);
    float hjv = hj[js];                          // LDS gather
    float rc  = fminf(rv, md);
    float d   = rc - r_peak;
    float g   = __expf(-d * d * inv2s2);         // v_exp_f32
    float valid = (rv < md - 1e-4f) ? 1.0f : 0.0f;
    return hjv * g * valid;
}

__global__ __launch_bounds__(THREADS)
void _HydrophobicPairs_kernel(const int* __restrict__ seq,
                              const float* __restrict__ r,
                              const int* __restrict__ jidx,
                              const float* __restrict__ h,
                              const float* __restrict__ rhalf_raw,
                              const float* __restrict__ tau_raw,
                              const int* __restrict__ maxd,
                              float* __restrict__ out) {
    __shared__ union { int i[LROW]; float f[LROW]; } sbuf;   // 32 KB

    const int tid = threadIdx.x;
    const int b   = blockIdx.x / TILES_PER_B;                // compile-time divisor -> shift
    const int l0  = (blockIdx.x % TILES_PER_B) * LTILE;

    // ---- Stage seq[b, 0:L] (32 KB) into LDS with CDNA5 async DMA ----
    {
        const unsigned long long base =
            (unsigned long long)(const void*)(seq + (size_t)b * LROW);
        const unsigned lds_base = (unsigned)(uintptr_t)(void*)sbuf.i;
        #pragma unroll
        for (int p = 0; p < (LROW * 4) / (THREADS * 16); ++p) {   // 8 passes, b128/lane
            unsigned off   = (unsigned)(p * THREADS * 16 + tid * 16);
            unsigned laddr = lds_base + off;
            asm volatile("global_load_async_to_lds_b128 %0, %1, %2"
                         :: "v"(laddr), "v"(off), "s"(base) : "memory");
        }
        asm volatile("s_wait_asynccnt 0" ::: "memory");
    }
    __syncthreads();

    // ---- In-place convert staged seq ints -> h[seq] floats ----
    {
        int tmp[LROW / THREADS];
        #pragma unroll
        for (int k = 0; k < LROW / THREADS; ++k)
            tmp[k] = sbuf.i[tid + k * THREADS];
        __syncthreads();
        #pragma unroll
        for (int k = 0; k < LROW / THREADS; ++k)
            sbuf.f[tid + k * THREADS] = h[tmp[k]];               // h is 20 floats, cache-hot
    }
    __syncthreads();

    // ---- Uniform scalars ----
    const float md     = (float)maxd[0];
    const float r_peak = log1pf(__expf(rhalf_raw[0]));           // softplus
    const float sigma  = log1pf(__expf(tau_raw[0])) + 0.1f;
    const float inv2s2 = 1.0f / (2.0f * sigma * sigma);

    // ---- Main reduction: one thread per output, nontemporal b128 streams ----
    #pragma unroll
    for (int ll = 0; ll < LTILE / THREADS; ++ll) {               // 2 outputs/thread
        const int l = l0 + ll * THREADS + tid;
        const size_t rowOff = ((size_t)b * LROW + l) * (size_t)KNB;
        const v4f* r4 = (const v4f*)(r + rowOff);
        const v4i* j4 = (const v4i*)(jidx + rowOff);
        float acc = 0.0f;
        #pragma unroll 4
        for (int q = 0; q < KNB / 4; ++q) {
            v4f rv = __builtin_nontemporal_load(r4 + q);         // th:NT, zero reuse stream
            v4i jv = __builtin_nontemporal_load(j4 + q);
            acc += hp_elem(rv.x, jv.x, sbuf.f, md, r_peak, inv2s2);
            acc += hp_elem(rv.y, jv.y, sbuf.f, md, r_peak, inv2s2);
            acc += hp_elem(rv.z, jv.z, sbuf.f, md, r_peak, inv2s2);
            acc += hp_elem(rv.w, jv.w, sbuf.f, md, r_peak, inv2s2);
        }
        out[(size_t)b * LROW + l] = sbuf.f[l] * acc;             // h_i free from staged row
    }
}

extern "C" void kernel_launch(void* const* d_in, const int* in_sizes, int n_in,
                              void* d_out, int out_size, void* d_ws, size_t ws_size,
                              hipStream_t stream) {
    const int*   seq   = (const int*)d_in[0];
    const float* r     = (const float*)d_in[1];
    const int*   jidx  = (const int*)d_in[2];
    const float* h     = (const float*)d_in[3];
    const float* rhalf = (const float*)d_in[4];
    const float* tau   = (const float*)d_in[5];
    const int*   maxd  = (const int*)d_in[6];
    float*       out   = (float*)d_out;

    const int B    = in_sizes[0] / LROW;       // 32
    const int grid = B * TILES_PER_B;          // 512 blocks
    _HydrophobicPairs_kernel<<<dim3(grid), dim3(THREADS), 0, stream>>>(
        seq, r, jidx, h, rhalf, tau, maxd, out);
}